// TemporalAttentionLayer_23218593202514
// MI455X (gfx1250) — compile-verified
//
#include <hip/hip_runtime.h>
#include <hip/hip_bf16.h>
#include <cstddef>

// ---------------------------------------------------------------------------
// TemporalAttentionLayer for MI455X (gfx1250, wave32, WMMA)
// B=16, K=128, F=128, C1=256, C2=128, C3=64
// ---------------------------------------------------------------------------

typedef __attribute__((ext_vector_type(2))) float v2f;
typedef __attribute__((ext_vector_type(8))) float v8f;

#define ALPHA_LRELU 0.2f

constexpr int Bv = 16;
constexpr int Kv = 128;
constexpr int Fv = 128;
constexpr int C1 = 256;   // 2*F
constexpr int C2 = 128;   // 2*F2
constexpr int C3 = 64;    // 2*F3
constexpr int F2v = 64;
constexpr int F3v = 32;

// ---------------------------------------------------------------------------
// f32 WMMA 16x32 strip:  D(16x32) = A(16 x kdim) * B(kdim x 32)
// One wave (32 lanes) per strip; the A fragment is reused for both 16x16
// N-tiles (2 WMMAs per A-load). Uses V_WMMA_F32_16X16X4_F32, K-step = 4.
// Fragment layout per CDNA5 ISA 7.12.2:
//   A 16x4:  lane m=lane&15, half kh=lane>>4; v0=A[m,k+2kh], v1=A[m,k+2kh+1]
//   B 4x16:  col=lane&15;   v0=B[k+2kh,col], v1=B[k+2kh+1,col]
//   C/D:     row = r + 8*kh, col = lane&15   (r = 0..7)
// ---------------------------------------------------------------------------
__device__ inline void wmma_strip_f32(const float* __restrict__ A, int lda,
                                      const float* __restrict__ Bm, int ldb,
                                      int kdim, v8f& c0, v8f& c1) {
    const int lane = threadIdx.x & 31;
    const int m    = lane & 15;
    const int kh   = (lane >> 4) & 1;
    for (int k = 0; k < kdim; k += 4) {
        v2f av, b0, b1;
        av.x = A[(size_t)m * lda + k + 2 * kh + 0];
        av.y = A[(size_t)m * lda + k + 2 * kh + 1];
        const float* Br0 = Bm + (size_t)(k + 2 * kh + 0) * ldb;
        const float* Br1 = Bm + (size_t)(k + 2 * kh + 1) * ldb;
        b0.x = Br0[m];       b0.y = Br1[m];
        b1.x = Br0[16 + m];  b1.y = Br1[16 + m];
        c0 = __builtin_amdgcn_wmma_f32_16x16x4_f32(
            false, av, false, b0, (short)0, c0, false, false);
        c1 = __builtin_amdgcn_wmma_f32_16x16x4_f32(
            false, av, false, b1, (short)0, c1, false, false);
    }
}

// out[b, m, n] = sum_f V[b, m, f] * W[f, n] (+ bias[n])
// grid = ((Krows/16)*(N/32), B), block = 32 (one wave / 16x32 strip)
__global__ void wmma_gemm_bias_kernel(const float* __restrict__ V,
                                      const float* __restrict__ W,
                                      const float* __restrict__ bias,
                                      float* __restrict__ out,
                                      int Krows, int Fin, int N, int ldw) {
    const int ntn  = N >> 5;
    const int tile = blockIdx.x;
    const int tm   = (tile / ntn) << 4;
    const int tn   = (tile % ntn) << 5;
    const int b    = blockIdx.y;

    const float* A  = V + (size_t)b * Krows * Fin + (size_t)tm * Fin;
    const float* Bm = W + tn;

    v8f c0 = {}, c1 = {};
    wmma_strip_f32(A, Fin, Bm, ldw, Fin, c0, c1);

    const int lane = threadIdx.x & 31;
    const int m    = lane & 15;
    const int kh   = (lane >> 4) & 1;
    const float bv0 = bias ? bias[tn + m]      : 0.0f;
    const float bv1 = bias ? bias[tn + 16 + m] : 0.0f;

    float* O = out + (size_t)b * Krows * N + (size_t)tm * N + tn;
    #pragma unroll
    for (int r = 0; r < 8; ++r) {
        float* Orow = O + (size_t)(r + 8 * kh) * N;
        Orow[m]      = c0[r] + bv0;
        Orow[16 + m] = c1[r] + bv1;
    }
}

// out[b, i, f] = sigmoid( sum_j att[b,i,j] * x[b,j,f] )
// grid = ((K/16)*(F/32), B), block = 32
__global__ void wmma_attn_sigmoid_kernel(const float* __restrict__ att,
                                         const float* __restrict__ x,
                                         float* __restrict__ out) {
    const int ntn  = Fv >> 5;   // 4
    const int tile = blockIdx.x;
    const int tm   = (tile / ntn) << 4;
    const int tn   = (tile % ntn) << 5;
    const int b    = blockIdx.y;

    const float* A  = att + (size_t)b * Kv * Kv + (size_t)tm * Kv;
    const float* Bm = x + (size_t)b * Kv * Fv + tn;

    v8f c0 = {}, c1 = {};
    wmma_strip_f32(A, Kv, Bm, Fv, Kv, c0, c1);

    const int lane = threadIdx.x & 31;
    const int m    = lane & 15;
    const int kh   = (lane >> 4) & 1;

    float* O = out + (size_t)b * Kv * Fv + (size_t)tm * Fv + tn;
    #pragma unroll
    for (int r = 0; r < 8; ++r) {
        float* Orow = O + (size_t)(r + 8 * kh) * Fv;
        Orow[m]      = 1.0f / (1.0f + __expf(-c0[r]));
        Orow[16 + m] = 1.0f / (1.0f + __expf(-c1[r]));
    }
}

// e[b,i,j] = sum_c a[c] * lrelu(hi[b,i,c] + hj[b,j,c])   (bias pre-folded in hi)
// grid = (K, B), block = K (=128). Dynamic LDS: 2*C floats (hi row + a).
__global__ void scores_kernel(const float* __restrict__ hi,
                              const float* __restrict__ hj,
                              const float* __restrict__ avec,
                              float* __restrict__ e, int C) {
    extern __shared__ float sm[];           // [0,C): hi row  [C,2C): a
    const int i = blockIdx.x;
    const int b = blockIdx.y;
    const int j = threadIdx.x;

    const float* hir = hi + ((size_t)b * Kv + i) * C;
    for (int c = j; c < C; c += Kv) {
        sm[c]     = hir[c];
        sm[C + c] = avec[c];
    }
    __syncthreads();

    const float* hjr = hj + ((size_t)b * Kv + j) * C;
    __builtin_prefetch(hjr, 0, 1);          // global_prefetch_b8
    float acc = 0.0f;
    for (int c = 0; c < C; c += 4) {
        __builtin_prefetch(hjr + c + 32, 0, 1);
        const float4 h4 = *reinterpret_cast<const float4*>(hjr + c);
        float z;
        z = sm[c + 0] + h4.x; acc = fmaf(sm[C + c + 0], (z >= 0.f ? z : ALPHA_LRELU * z), acc);
        z = sm[c + 1] + h4.y; acc = fmaf(sm[C + c + 1], (z >= 0.f ? z : ALPHA_LRELU * z), acc);
        z = sm[c + 2] + h4.z; acc = fmaf(sm[C + c + 2], (z >= 0.f ? z : ALPHA_LRELU * z), acc);
        z = sm[c + 3] + h4.w; acc = fmaf(sm[C + c + 3], (z >= 0.f ? z : ALPHA_LRELU * z), acc);
    }
    e[((size_t)b * Kv + i) * Kv + j] = acc;
}

// ---- block reductions over 128 threads (4 wave32) ----
__device__ inline float block_reduce_max(float v, float* red, int j) {
    red[j] = v; __syncthreads();
    for (int s = 64; s > 0; s >>= 1) {
        if (j < s) red[j] = fmaxf(red[j], red[j + s]);
        __syncthreads();
    }
    const float r = red[0]; __syncthreads();
    return r;
}
__device__ inline float block_reduce_sum(float v, float* red, int j) {
    red[j] = v; __syncthreads();
    for (int s = 64; s > 0; s >>= 1) {
        if (j < s) red[j] = red[j] + red[j + s];
        __syncthreads();
    }
    const float r = red[0]; __syncthreads();
    return r;
}

// softmax(e1/e2/e3 along j) -> att = lrelu(p1*w0+p2*w1+p3*w2+b) [* wm]
// grid = (K, B), block = K
__global__ void softmax_combine_kernel(const float* __restrict__ e1,
                                       const float* __restrict__ e2,
                                       const float* __restrict__ e3,
                                       const float* __restrict__ mul_w,
                                       const float* __restrict__ mul_b,
                                       const float* __restrict__ wm,
                                       float* __restrict__ att, int apply_wm) {
    __shared__ float red[Kv];
    const int i = blockIdx.x;
    const int b = blockIdx.y;
    const int j = threadIdx.x;
    const size_t row = ((size_t)b * Kv + i) * Kv;

    const float v1 = e1[row + j];
    const float v2 = e2[row + j];
    const float v3 = e3[row + j];

    const float m1 = block_reduce_max(v1, red, j);
    const float m2 = block_reduce_max(v2, red, j);
    const float m3 = block_reduce_max(v3, red, j);

    const float p1 = __expf(v1 - m1);
    const float p2 = __expf(v2 - m2);
    const float p3 = __expf(v3 - m3);

    const float s1 = block_reduce_sum(p1, red, j);
    const float s2 = block_reduce_sum(p2, red, j);
    const float s3 = block_reduce_sum(p3, red, j);

    float o = (p1 / s1) * mul_w[0] + (p2 / s2) * mul_w[1] + (p3 / s3) * mul_w[2] + mul_b[0];
    o = (o >= 0.f) ? o : ALPHA_LRELU * o;
    if (apply_wm) o *= wm[row + j];
    att[row + j] = o;
}

// ---------------------------------------------------------------------------
extern "C" void kernel_launch(void* const* d_in, const int* in_sizes, int n_in,
                              void* d_out, int out_size, void* d_ws, size_t ws_size,
                              hipStream_t stream) {
    (void)in_sizes; (void)n_in; (void)out_size; (void)ws_size;

    const float* x        = (const float*)d_in[0];
    const float* dif_x    = (const float*)d_in[1];
    const float* wm       = (const float*)d_in[2];
    const float* lin_w    = (const float*)d_in[3];
    const float* lin_b    = (const float*)d_in[4];
    const float* a1       = (const float*)d_in[5];
    const float* embed2_w = (const float*)d_in[6];
    const float* embed2_b = (const float*)d_in[7];
    const float* lin2_w   = (const float*)d_in[8];
    const float* lin2_b   = (const float*)d_in[9];
    const float* a2       = (const float*)d_in[10];
    const float* embed3_w = (const float*)d_in[11];
    const float* embed3_b = (const float*)d_in[12];
    const float* lin3_w   = (const float*)d_in[13];
    const float* lin3_b   = (const float*)d_in[14];
    const float* a3       = (const float*)d_in[15];
    const float* mul_w    = (const float*)d_in[16];
    const float* mul_b    = (const float*)d_in[17];
    float* out = (float*)d_out;

    // workspace carve-up (floats), reused across the two branches
    float* ws  = (float*)d_ws;
    float* hi1 = ws;                 ws += (size_t)Bv * Kv * C1;
    float* hj1 = ws;                 ws += (size_t)Bv * Kv * C1;
    float* v2b = ws;                 ws += (size_t)Bv * Kv * F2v;
    float* v3b = ws;                 ws += (size_t)Bv * Kv * F3v;
    float* hi2 = ws;                 ws += (size_t)Bv * Kv * C2;
    float* hj2 = ws;                 ws += (size_t)Bv * Kv * C2;
    float* hi3 = ws;                 ws += (size_t)Bv * Kv * C3;
    float* hj3 = ws;                 ws += (size_t)Bv * Kv * C3;
    float* e1  = ws;                 ws += (size_t)Bv * Kv * Kv;
    float* e2  = ws;                 ws += (size_t)Bv * Kv * Kv;
    float* e3  = ws;                 ws += (size_t)Bv * Kv * Kv;
    float* att = ws;                 ws += (size_t)Bv * Kv * Kv;

    const dim3 wave(32);
    const dim3 rowsBlk(Kv);
    const dim3 rowsGrid(Kv, Bv);

    for (int br = 0; br < 2; ++br) {
        const float* v = (br == 0) ? x : dif_x;

        // hi1 = v @ lin_w[:F] + lin_b ; hj1 = v @ lin_w[F:]
        dim3 g1((Kv / 16) * (C1 / 32), Bv);
        wmma_gemm_bias_kernel<<<g1, wave, 0, stream>>>(v, lin_w,            lin_b,   hi1, Kv, Fv, C1, C1);
        wmma_gemm_bias_kernel<<<g1, wave, 0, stream>>>(v, lin_w + Fv * C1,  nullptr, hj1, Kv, Fv, C1, C1);

        // v2 = v @ embed2_w + embed2_b ; v3 = v @ embed3_w + embed3_b
        dim3 g2((Kv / 16) * (F2v / 32), Bv);
        wmma_gemm_bias_kernel<<<g2, wave, 0, stream>>>(v, embed2_w, embed2_b, v2b, Kv, Fv, F2v, F2v);
        dim3 g3((Kv / 16) * (F3v / 32), Bv);
        wmma_gemm_bias_kernel<<<g3, wave, 0, stream>>>(v, embed3_w, embed3_b, v3b, Kv, Fv, F3v, F3v);

        // hi2/hj2 from v2, hi3/hj3 from v3
        dim3 g4((Kv / 16) * (C2 / 32), Bv);
        wmma_gemm_bias_kernel<<<g4, wave, 0, stream>>>(v2b, lin2_w,             lin2_b,  hi2, Kv, F2v, C2, C2);
        wmma_gemm_bias_kernel<<<g4, wave, 0, stream>>>(v2b, lin2_w + F2v * C2,  nullptr, hj2, Kv, F2v, C2, C2);
        dim3 g5((Kv / 16) * (C3 / 32), Bv);
        wmma_gemm_bias_kernel<<<g5, wave, 0, stream>>>(v3b, lin3_w,             lin3_b,  hi3, Kv, F3v, C3, C3);
        wmma_gemm_bias_kernel<<<g5, wave, 0, stream>>>(v3b, lin3_w + F3v * C3,  nullptr, hj3, Kv, F3v, C3, C3);

        // dense GAT scores (lrelu inside the channel reduction -> VALU kernel)
        scores_kernel<<<rowsGrid, rowsBlk, 2 * C1 * sizeof(float), stream>>>(hi1, hj1, a1, e1, C1);
        scores_kernel<<<rowsGrid, rowsBlk, 2 * C2 * sizeof(float), stream>>>(hi2, hj2, a2, e2, C2);
        scores_kernel<<<rowsGrid, rowsBlk, 2 * C3 * sizeof(float), stream>>>(hi3, hj3, a3, e3, C3);

        // softmax over j (axis=2), 3-way mix + lrelu, ts-branch scaled by wm
        softmax_combine_kernel<<<rowsGrid, rowsBlk, 0, stream>>>(
            e1, e2, e3, mul_w, mul_b, wm, att, (br == 0) ? 1 : 0);

        // h = sigmoid(att @ x)  (both branches multiply by x per reference)
        dim3 gf((Kv / 16) * (Fv / 32), Bv);
        wmma_attn_sigmoid_kernel<<<gf, wave, 0, stream>>>(
            att, x, out + (size_t)br * Bv * Kv * Fv);
    }
}